// SingleAttentionAggregator_72164040508246
// MI455X (gfx1250) — compile-verified
//
#include <hip/hip_runtime.h>
#include <cstddef>

// Problem constants (match reference setup_inputs()).
#define NROWS 100000      // destination nodes
#define MROWS 100000      // source nodes
#define NEDGE 3200000     // edges
#define DIM   128
#define LEAKY 0.2f

typedef __attribute__((ext_vector_type(2))) float v2f;
typedef __attribute__((ext_vector_type(8))) float v8f;

// ---- monotonic uint key for float atomic max (order independent => deterministic) ----
__device__ __forceinline__ unsigned fkey(float f) {
    unsigned u = __float_as_uint(f);
    return (u & 0x80000000u) ? ~u : (u | 0x80000000u);
}
__device__ __forceinline__ float fdec(unsigned k) {
    return (k & 0x80000000u) ? __uint_as_float(k & 0x7FFFFFFFu)
                             : __uint_as_float(~k);
}

// ---------------------------------------------------------------------------
// Transpose W [128x128] -> Wt (column major) so WMMA B fragments are float2 loads.
__global__ __launch_bounds__(256) void k_transpose_w(const float* __restrict__ W,
                                                     float* __restrict__ Wt) {
    int idx = blockIdx.x * 256 + threadIdx.x;   // 16384 elements
    int n = idx >> 7;
    int k = idx & 127;
    Wt[idx] = W[k * DIM + n];                   // Wt[n*128 + k] = W[k,n]
}

// ---------------------------------------------------------------------------
// Projection: Y = X @ W (via Wt), fused logit[r] = dot(Y[r,:], wvec).
// Block = 256 threads = 8 waves; block computes 16 rows x 128 cols.
// Wave w computes the 16x16 tile for cols [16w, 16w+16) using
// V_WMMA_F32_16X16X4_F32, K-loop of 32 steps (K=128).
__global__ __launch_bounds__(256) void k_proj(const float* __restrict__ X,
                                              const float* __restrict__ Wt,
                                              const float* __restrict__ wvec,
                                              float* __restrict__ Y,
                                              float* __restrict__ logit) {
    __shared__ float red[16][8];

    const int wave = threadIdx.x >> 5;
    const int lane = threadIdx.x & 31;
    const int hl   = lane >> 4;        // which 16-lane half
    const int l16  = lane & 15;
    const int row0 = blockIdx.x * 16;
    const int col  = wave * 16 + l16;  // global output column
    const int koff = hl * 2;           // A/B K sub-offset per ISA layout

    const float* __restrict__ xrow = X  + (size_t)(row0 + l16) * DIM + koff;
    const float* __restrict__ wcol = Wt + (size_t)col * DIM + koff;

    v8f acc = {};
#pragma unroll
    for (int k0 = 0; k0 < DIM; k0 += 4) {
        v2f a = *reinterpret_cast<const v2f*>(xrow + k0);  // X[row, k0+koff .. +1]
        v2f b = *reinterpret_cast<const v2f*>(wcol + k0);  // W[k0+koff .. +1, col]
        acc = __builtin_amdgcn_wmma_f32_16x16x4_f32(
            /*neg_a=*/false, a, /*neg_b=*/false, b,
            /*c_mod=*/(short)0, acc, /*reuse_a=*/false, /*reuse_b=*/false);
    }

    const float wvc = wvec[col];
#pragma unroll
    for (int v = 0; v < 8; ++v) {
        const int r = row0 + v + 8 * hl;          // C/D layout: M = v + 8*half
        Y[(size_t)r * DIM + col] = acc[v];
        float s = acc[v] * wvc;                   // partial dot over this col tile
#pragma unroll
        for (int off = 1; off < 16; off <<= 1)    // butterfly within 16-lane half
            s += __shfl_xor(s, off, 32);
        if (l16 == 0) red[v + 8 * hl][wave] = s;
    }
    __syncthreads();
    if (threadIdx.x < 16) {
        float t = 0.f;
#pragma unroll
        for (int w = 0; w < 8; ++w) t += red[threadIdx.x][w];
        logit[row0 + threadIdx.x] = t;
    }
}

// ---------------------------------------------------------------------------
// Pass 1 over edges: score, segment-max (monotone-key atomicMax), degree histogram.
__global__ __launch_bounds__(256) void k_score(const int* __restrict__ erow,
                                               const int* __restrict__ ecol,
                                               const float* __restrict__ self_w,
                                               const float* __restrict__ all_w,
                                               float* __restrict__ scores,
                                               unsigned* __restrict__ rowmax,
                                               unsigned* __restrict__ cnt) {
    int e = blockIdx.x * 256 + threadIdx.x;
    if (e >= NEDGE) return;
    int r = erow[e], c = ecol[e];
    float x = self_w[r] + all_w[c];
    float s = x > 0.f ? x : LEAKY * x;
    scores[e] = s;
    atomicMax(&rowmax[r], fkey(s));
    atomicAdd(&cnt[r], 1u);
}

// ---------------------------------------------------------------------------
// Two-level exclusive scan of per-row counts (N = 100000, tile = 1024).
__global__ __launch_bounds__(256) void k_scan1(const unsigned* __restrict__ cnt,
                                               unsigned* __restrict__ out,
                                               unsigned* __restrict__ bsum, int n) {
    __shared__ unsigned sh[256];
    const int t = threadIdx.x;
    const int base = blockIdx.x * 1024 + t * 4;
    unsigned v[4];
#pragma unroll
    for (int j = 0; j < 4; ++j) v[j] = (base + j < n) ? cnt[base + j] : 0u;
    unsigned tsum = v[0] + v[1] + v[2] + v[3];
    sh[t] = tsum;
    __syncthreads();
    for (int off = 1; off < 256; off <<= 1) {
        unsigned u = (t >= off) ? sh[t - off] : 0u;
        __syncthreads();
        sh[t] += u;
        __syncthreads();
    }
    unsigned excl = sh[t] - tsum;
#pragma unroll
    for (int j = 0; j < 4; ++j) {
        if (base + j < n) out[base + j] = excl;
        excl += v[j];
    }
    if (t == 255) bsum[blockIdx.x] = sh[255];
}

__global__ __launch_bounds__(256) void k_scan2(unsigned* __restrict__ bsum, int nb) {
    __shared__ unsigned sh[256];
    const int t = threadIdx.x;
    unsigned v = (t < nb) ? bsum[t] : 0u;
    sh[t] = v;
    __syncthreads();
    for (int off = 1; off < 256; off <<= 1) {
        unsigned u = (t >= off) ? sh[t - off] : 0u;
        __syncthreads();
        sh[t] += u;
        __syncthreads();
    }
    if (t < nb) bsum[t] = sh[t] - v;   // exclusive
}

__global__ __launch_bounds__(256) void k_scan3(unsigned* __restrict__ row_start,
                                               unsigned* __restrict__ cursor,
                                               const unsigned* __restrict__ bsum, int n) {
    int i = blockIdx.x * 256 + threadIdx.x;
    if (i >= n) return;
    unsigned v = row_start[i] + bsum[i >> 10];
    row_start[i] = v;
    cursor[i]    = v;
}

// ---------------------------------------------------------------------------
// Pass 2 over edges: e = exp(score - rowmax), scatter (col, e) into CSR slots.
__global__ __launch_bounds__(256) void k_scatter(const int* __restrict__ erow,
                                                 const int* __restrict__ ecol,
                                                 const float* __restrict__ scores,
                                                 const unsigned* __restrict__ rowmax,
                                                 unsigned* __restrict__ cursor,
                                                 unsigned* __restrict__ perm_col,
                                                 float* __restrict__ perm_w) {
    int e = blockIdx.x * 256 + threadIdx.x;
    if (e >= NEDGE) return;
    int r = erow[e];
    float ev = __expf(scores[e] - fdec(rowmax[r]));
    unsigned pos = atomicAdd(&cursor[r], 1u);
    perm_col[pos] = (unsigned)ecol[e];
    perm_w[pos]   = ev;
}

// ---------------------------------------------------------------------------
// Aggregation: one wave per row. Lane owns cols [4*lane, 4*lane+4) (float4).
// Accumulates Sum(e * from_all[col]) and the softmax denominator in-register,
// then out = relu(from_self + acc/denom).  from_all (51 MB) is L2-resident.
__global__ __launch_bounds__(256) void k_agg(const float* __restrict__ from_all,
                                             const float* __restrict__ from_self,
                                             const unsigned* __restrict__ row_start,
                                             const unsigned* __restrict__ cnt,
                                             const unsigned* __restrict__ perm_col,
                                             const float* __restrict__ perm_w,
                                             float* __restrict__ out) {
    const int wave = threadIdx.x >> 5;
    const int lane = threadIdx.x & 31;
    const int i = blockIdx.x * 8 + wave;
    if (i >= NROWS) return;

    const unsigned start = row_start[i];
    const unsigned n     = cnt[i];

    float4 acc = make_float4(0.f, 0.f, 0.f, 0.f);
    float  den = 0.f;
    for (unsigned j = 0; j < n; ++j) {
        unsigned c = perm_col[start + j];           // uniform across wave
        float    w = perm_w[start + j];
        den += w;
        const float4 v = *reinterpret_cast<const float4*>(
            from_all + (size_t)c * DIM + lane * 4);
        acc.x += w * v.x; acc.y += w * v.y; acc.z += w * v.z; acc.w += w * v.w;
    }
    const float scale = den > 0.f ? 1.f / den : 0.f;
    const float4 fs = *reinterpret_cast<const float4*>(
        from_self + (size_t)i * DIM + lane * 4);
    float4 o;
    o.x = fmaxf(fs.x + acc.x * scale, 0.f);
    o.y = fmaxf(fs.y + acc.y * scale, 0.f);
    o.z = fmaxf(fs.z + acc.z * scale, 0.f);
    o.w = fmaxf(fs.w + acc.w * scale, 0.f);
    *reinterpret_cast<float4*>(out + (size_t)i * DIM + lane * 4) = o;
}

// ---------------------------------------------------------------------------
extern "C" void kernel_launch(void* const* d_in, const int* in_sizes, int n_in,
                              void* d_out, int out_size, void* d_ws, size_t ws_size,
                              hipStream_t stream) {
    const float* self_emb  = (const float*)d_in[0];   // [N,128]
    const float* neigh_emb = (const float*)d_in[1];   // [M,128]
    const float* W         = (const float*)d_in[2];   // [128,128]
    const float* w_self    = (const float*)d_in[3];   // [128]
    const float* w_neigh   = (const float*)d_in[4];   // [128]
    const int*   erow      = (const int*)d_in[5];     // [E]
    const int*   ecol      = (const int*)d_in[6];     // [E]
    float* out = (float*)d_out;                       // [N,128]

    // Workspace carve-up (256B-aligned), ~144 MB total.
    size_t off = 0;
    auto carve = [&](size_t bytes) -> void* {
        void* p = (char*)d_ws + off;
        off += (bytes + 255) & ~(size_t)255;
        return p;
    };
    float*    from_self = (float*)carve((size_t)NROWS * DIM * 4);
    float*    from_all  = (float*)carve((size_t)MROWS * DIM * 4);
    float*    Wt        = (float*)carve((size_t)DIM * DIM * 4);
    float*    self_w    = (float*)carve((size_t)NROWS * 4);
    float*    all_w     = (float*)carve((size_t)MROWS * 4);
    float*    scores    = (float*)carve((size_t)NEDGE * 4);
    unsigned* rowmax    = (unsigned*)carve((size_t)NROWS * 4);
    unsigned* cnt       = (unsigned*)carve((size_t)NROWS * 4);
    unsigned* row_start = (unsigned*)carve((size_t)NROWS * 4);
    unsigned* cursor    = (unsigned*)carve((size_t)NROWS * 4);
    unsigned* bsum      = (unsigned*)carve(256 * 4);
    unsigned* perm_col  = (unsigned*)carve((size_t)NEDGE * 4);
    float*    perm_w    = (float*)carve((size_t)NEDGE * 4);

    hipMemsetAsync(rowmax, 0, (size_t)NROWS * 4, stream); // fkey sentinel (= -inf)
    hipMemsetAsync(cnt,    0, (size_t)NROWS * 4, stream);

    // 1) W transpose for coalesced WMMA B fragments.
    k_transpose_w<<<(DIM * DIM) / 256, 256, 0, stream>>>(W, Wt);

    // 2) Dense projections + fused attention logits (f32 WMMA).
    k_proj<<<NROWS / 16, 256, 0, stream>>>(self_emb,  Wt, w_self,  from_self, self_w);
    k_proj<<<MROWS / 16, 256, 0, stream>>>(neigh_emb, Wt, w_neigh, from_all,  all_w);

    // 3) Edge scores + segment max + degree histogram.
    const int eblocks = (NEDGE + 255) / 256;
    k_score<<<eblocks, 256, 0, stream>>>(erow, ecol, self_w, all_w,
                                         scores, rowmax, cnt);

    // 4) Exclusive scan of degrees -> CSR row starts (+ cursor copy).
    const int nb = (NROWS + 1023) / 1024;       // 98
    k_scan1<<<nb, 256, 0, stream>>>(cnt, row_start, bsum, NROWS);
    k_scan2<<<1, 256, 0, stream>>>(bsum, nb);
    k_scan3<<<(NROWS + 255) / 256, 256, 0, stream>>>(row_start, cursor, bsum, NROWS);

    // 5) exp(score - max) + CSR scatter.
    k_scatter<<<eblocks, 256, 0, stream>>>(erow, ecol, scores, rowmax,
                                           cursor, perm_col, perm_w);

    // 6) Per-row wave aggregation, fused softmax-normalize + residual + relu.
    k_agg<<<NROWS / 8, 256, 0, stream>>>(from_all, from_self, row_start, cnt,
                                         perm_col, perm_w, out);
}